// SamplingBlock_36137854828694
// MI455X (gfx1250) — compile-verified
//
#include <hip/hip_runtime.h>
#include <math.h>

// Stochastic binomial amplitude quantization for x: (32,4096,1024) fp32.
//   pass1: norm2 = sum(x^2)   (WMMA 16x16x4_f32 as reduction engine, guard-free hot loop)
//   pass2: m_i ~ Binom(97200, x_i^2/norm2), out_i = sign(x_i)*sqrt(m_i), accumulate sum(m)
//   pass3: out *= ||y|| / ||m||, with lo/hi norm guards (uniform-branch hoisted).

typedef __attribute__((ext_vector_type(2))) float v2f;
typedef __attribute__((ext_vector_type(8))) float v8f;

#define N_TRIALS_F 97200.0f   // int(36 * log2(2^27) / 0.1^2)
#define NORM_LO2   1e-16      // (1e-8)^2
#define NORM_HI2   1e26       // (1e13)^2

__device__ __forceinline__ float waveReduceSum(float v) {
  #pragma unroll
  for (int off = 16; off >= 1; off >>= 1) v += __shfl_xor(v, off, 32);  // wave32
  return v;
}

__device__ __forceinline__ unsigned long long mix64(unsigned long long z) {
  z += 0x9E3779B97F4A7C15ull;
  z = (z ^ (z >> 30)) * 0xBF58476D1CE4E5B9ull;
  z = (z ^ (z >> 27)) * 0x94D049BB133111EBull;
  return z ^ (z >> 31);
}

__device__ __forceinline__ float u01(unsigned long long& st) {
  st = st * 6364136223846793005ull + 1442695040888963407ull;
  unsigned r = (unsigned)(st >> 40);                     // 24 random bits
  return ((float)r + 0.5f) * (1.0f / 16777216.0f);       // (0,1)
}

__device__ __forceinline__ v2f sq_pair(float4 v) {
  v2f a;
  a.x = fmaf(v.x, v.x, v.z * v.z);
  a.y = fmaf(v.y, v.y, v.w * v.w);
  return a;
}

// ---------------- Pass 1: sum of squares via WMMA row-sum reduction ----------------
__global__ void norm2_wmma_kernel(const float* __restrict__ x, int n,
                                  double* __restrict__ acc_norm2) {
  const int tid      = blockIdx.x * blockDim.x + threadIdx.x;
  const int nthreads = gridDim.x * blockDim.x;
  const int n4       = n >> 2;
  const int span     = nthreads * 4;        // float4s consumed per unrolled iter
  const int main_it  = n4 / span;           // grid-uniform trip count: no guards,
                                            // EXEC stays all-1s for every WMMA
  const float4* __restrict__ xp = reinterpret_cast<const float4*>(x);

  v2f ones; ones.x = 1.0f; ones.y = 1.0f;   // B = ones(4x16) -> D rows = row-sums of A
  v8f c0 = {}, c1 = {}, c2 = {}, c3 = {};   // 4 independent accumulators

  for (int it = 0; it < main_it; ++it) {
    const int b = it * span + tid;
    // 4 independent b128 loads -> memory-level parallelism
    float4 v0 = xp[b];
    float4 v1 = xp[b + nthreads];
    float4 v2 = xp[b + 2 * nthreads];
    float4 v3 = xp[b + 3 * nthreads];
    v2f a0 = sq_pair(v0), a1 = sq_pair(v1), a2 = sq_pair(v2), a3 = sq_pair(v3);
    c0 = __builtin_amdgcn_wmma_f32_16x16x4_f32(false, a0, false, ones, (short)0, c0, false, false);
    c1 = __builtin_amdgcn_wmma_f32_16x16x4_f32(false, a1, false, ones, (short)0, c1, false, false);
    c2 = __builtin_amdgcn_wmma_f32_16x16x4_f32(false, a2, false, ones, (short)0, c2, false, false);
    c3 = __builtin_amdgcn_wmma_f32_16x16x4_f32(false, a3, false, ones, (short)0, c3, false, false);
  }

  // Tail (guarded, WMMA-free): leftover float4s, then leftover scalars.
  float tail = 0.0f;
  for (int idx = main_it * span + tid; idx < n4; idx += nthreads) {
    float4 v = xp[idx];
    tail += fmaf(v.x, v.x, v.y * v.y) + fmaf(v.z, v.z, v.w * v.w);
  }
  for (int e = (n4 << 2) + tid; e < n; e += nthreads) tail += x[e] * x[e];

  // All 16 D columns identical -> waveSum over all entries = 16 * total.
  float s = 0.0f;
  #pragma unroll
  for (int k = 0; k < 8; ++k) s += (c0[k] + c1[k]) + (c2[k] + c3[k]);
  s = s * (1.0f / 16.0f) + tail;
  s = waveReduceSum(s);
  if ((threadIdx.x & 31) == 0) atomicAdd(acc_norm2, (double)s);
}

// ---------------- Pass 2: binomial sampling + unnormalized output ----------------
__device__ __forceinline__ float sample_binomial(float xe, float inv_norm2,
                                                 unsigned long long st) {
  float p = fminf(xe * xe * inv_norm2, 1.0f);
  float lam = N_TRIALS_F * p;
  float m;
  if (lam < 10.0f) {
    // Knuth Poisson (lam ~ 7e-4 on average: expected 1 iteration)
    float L = __expf(-lam);
    float t = 1.0f;
    int k = 0;
    #pragma unroll 1
    do {
      ++k;
      t *= u01(st);
    } while (t > L && k < 64);
    m = (float)(k - 1);
  } else {
    // Gaussian approx of Binomial(n,p): mean np, var np(1-p)
    float u1 = u01(st), u2 = u01(st);
    float z = sqrtf(fmaxf(-2.0f * __logf(u1), 0.0f)) * __cosf(6.2831853f * u2);
    m = rintf(lam + sqrtf(lam * (1.0f - p)) * z);
    m = fminf(fmaxf(m, 0.0f), N_TRIALS_F);
  }
  return m;
}

__global__ void quantize_kernel(const float* __restrict__ x, float* __restrict__ out,
                                int n, const double* __restrict__ acc_norm2,
                                double* __restrict__ acc_summ) {
  const float norm2 = (float)acc_norm2[0];
  const float inv_norm2 = (norm2 > 0.0f) ? (1.0f / norm2) : 0.0f;

  const int tid      = blockIdx.x * blockDim.x + threadIdx.x;
  const int nthreads = gridDim.x * blockDim.x;
  const int n4       = n >> 2;
  const float4* __restrict__ xp = reinterpret_cast<const float4*>(x);
  float4* __restrict__ op = reinterpret_cast<float4*>(out);

  float local_sum = 0.0f;
  for (int v = tid; v < n4; v += nthreads) {
    if (v + nthreads < n4) __builtin_prefetch(&xp[v + nthreads], 0, 1);
    float4 xv = xp[v];
    const unsigned long long e0 = (unsigned long long)v << 2;
    float m0 = sample_binomial(xv.x, inv_norm2, mix64(e0 + 0));
    float m1 = sample_binomial(xv.y, inv_norm2, mix64(e0 + 1));
    float m2 = sample_binomial(xv.z, inv_norm2, mix64(e0 + 2));
    float m3 = sample_binomial(xv.w, inv_norm2, mix64(e0 + 3));
    local_sum += (m0 + m1) + (m2 + m3);
    float4 ov;
    ov.x = copysignf(sqrtf(m0), xv.x);
    ov.y = copysignf(sqrtf(m1), xv.y);
    ov.z = copysignf(sqrtf(m2), xv.z);
    ov.w = copysignf(sqrtf(m3), xv.w);
    op[v] = ov;
  }
  // scalar tail
  for (int e = (n4 << 2) + tid; e < n; e += nthreads) {
    float xe = x[e];
    float m = sample_binomial(xe, inv_norm2, mix64((unsigned long long)e));
    local_sum += m;
    out[e] = copysignf(sqrtf(m), xe);
  }

  local_sum = waveReduceSum(local_sum);
  if ((threadIdx.x & 31) == 0) atomicAdd(acc_summ, (double)local_sum);
}

// ---------------- Pass 3: renormalize + guards (uniform branches hoisted) ----------------
__global__ void renorm_kernel(const float* __restrict__ x, float* __restrict__ out,
                              int n, const double* __restrict__ ws) {
  const double norm2 = ws[0];
  const double summ  = ws[1];   // ||m_vec||^2 == sum(m) since elements are sqrt(m)
  const bool hi_pass = (norm2 >= NORM_HI2);   // y_norm >= 1e13 -> passthrough x (dominates)
  const bool lo_zero = !(norm2 > NORM_LO2);   // y_norm <= 1e-8 -> zeros
  const float scale  = (summ > 0.0) ? (float)sqrt(norm2 / summ) : 0.0f;

  const int tid      = blockIdx.x * blockDim.x + threadIdx.x;
  const int nthreads = gridDim.x * blockDim.x;
  const int n4       = n >> 2;
  const float4* __restrict__ xp = reinterpret_cast<const float4*>(x);
  float4* __restrict__ op = reinterpret_cast<float4*>(out);

  if (hi_pass) {
    for (int v = tid; v < n4; v += nthreads) op[v] = xp[v];
    for (int e = (n4 << 2) + tid; e < n; e += nthreads) out[e] = x[e];
  } else if (lo_zero) {
    float4 z; z.x = z.y = z.z = z.w = 0.0f;
    for (int v = tid; v < n4; v += nthreads) op[v] = z;
    for (int e = (n4 << 2) + tid; e < n; e += nthreads) out[e] = 0.0f;
  } else {
    for (int v = tid; v < n4; v += nthreads) {
      float4 t = op[v];
      t.x *= scale; t.y *= scale; t.z *= scale; t.w *= scale;
      op[v] = t;
    }
    for (int e = (n4 << 2) + tid; e < n; e += nthreads) out[e] *= scale;
  }
}

extern "C" void kernel_launch(void* const* d_in, const int* in_sizes, int n_in,
                              void* d_out, int out_size, void* d_ws, size_t ws_size,
                              hipStream_t stream) {
  const float* x = (const float*)d_in[0];
  float* out = (float*)d_out;
  const int n = in_sizes[0];   // 134,217,728 (fits int32)

  double* acc = (double*)d_ws;  // acc[0] = norm2, acc[1] = sum(m)
  hipMemsetAsync(d_ws, 0, 2 * sizeof(double), stream);

  dim3 blk(256);  // 8 waves/block on wave32
  norm2_wmma_kernel<<<dim3(8192), blk, 0, stream>>>(x, n, acc);
  quantize_kernel<<<dim3(16384), blk, 0, stream>>>(x, out, n, acc, acc + 1);
  renorm_kernel<<<dim3(8192), blk, 0, stream>>>(x, out, n, acc);
}